// NeuralODE_84232898609513
// MI455X (gfx1250) — compile-verified
//
#include <hip/hip_runtime.h>

typedef __attribute__((ext_vector_type(2))) float v2f;
typedef __attribute__((ext_vector_type(8))) float v8f;

// Neural ODE, explicit Euler, MLP 1 -> 16 -> 16 -> 1 with ReLU.
// One wave32 handles 2 tiles of 16 trajectories (32 trajectories/wave).
// Layer 2 (16x16 matvec batched over 16 trajectories) runs on
// v_wmma_f32_16x16x4_f32 chained over K=16, computing H2^T = W2^T x H1^T so
// that the trajectory index stays on lanes and no transpose is ever needed.
__global__ __launch_bounds__(256) void node_euler_wmma(
    const float* __restrict__ y0, const float* __restrict__ tgrid,
    const float* __restrict__ W1, const float* __restrict__ b1,
    const float* __restrict__ W2, const float* __restrict__ b2,
    const float* __restrict__ W3, const float* __restrict__ b3,
    float* __restrict__ out, int B, int T)
{
    const int lane = threadIdx.x & 31;
    const int wave = threadIdx.x >> 5;
    const int hi   = lane >> 4;   // lane half (0: lanes 0-15, 1: lanes 16-31)
    const int l16  = lane & 15;   // trajectory-within-tile index
    const int base = (blockIdx.x * 8 + wave) * 32;  // 32 trajectories per wave

    const float inf = __builtin_inff();

    // ---- Load lane-resident constants (once) -------------------------------
    // A-operand of layer-2 WMMA: W2^T in 32-bit A layout.
    //   reg i holds K = 4*(i/2) + (i%2) + 2*hi, M = l16  ->  W2[K][l16].
    // Same K mapping gives W1/b1 values for building H1^T in B layout
    // (B layout: reg i holds K = same formula, N = l16 = trajectory).
    float w2A[8], w1a[8], b1a[8];
#pragma unroll
    for (int i = 0; i < 8; ++i) {
        const int kd = ((i >> 1) << 2) + (i & 1) + (hi << 1);
        w2A[i] = W2[kd * 16 + l16];
        w1a[i] = W1[kd];
        b1a[i] = b1[kd];
    }
    // C/D layout: reg r, lane half hi  ->  hidden index r + 8*hi (rows of H2^T),
    // trajectory = l16 (columns). Bias b2 preloaded as the WMMA C operand.
    float w3c[8];
    v8f   b2c;
#pragma unroll
    for (int r = 0; r < 8; ++r) {
        w3c[r] = W3[r + 8 * hi];
        b2c[r] = b2[r + 8 * hi];
    }
    const float b3s = b3[0];

    // ---- Initial state: each lane owns trajectory (base + tau*16 + l16) ----
    float yA[2];
    yA[0] = y0[base + l16];
    yA[1] = y0[base + 16 + l16];

    // Row 0 of the output is y0 itself. Lanes 0-15 store tile 0, 16-31 tile 1:
    // one fully coalesced 128B store per wave.
    float* orow = out + base;
    orow[lane] = (lane < 16) ? yA[0] : yA[1];

    float tp = tgrid[0];
    for (int t = 1; t < T; ++t) {
        const float tc = tgrid[t];     // uniform scalar load (s_load_b32)
        const float dt = tc - tp;      // s_sub_f32
        tp = tc;
        orow += B;

#pragma unroll
        for (int tau = 0; tau < 2; ++tau) {
            // Layer 1: H1^T built directly in B layout (no data movement).
            float h[8];
#pragma unroll
            for (int i = 0; i < 8; ++i)
                h[i] = fmaxf(fmaf(yA[tau], w1a[i], b1a[i]), 0.0f);

            // Layer 2: H2^T = W2^T x H1^T + b2, K=16 as 4 chained
            // v_wmma_f32_16x16x4_f32 accumulations (bias fused via C input).
            v8f acc = b2c;
#pragma unroll
            for (int k = 0; k < 4; ++k) {
                v2f av; av.x = w2A[2 * k]; av.y = w2A[2 * k + 1];
                v2f bv; bv.x = h[2 * k];   bv.y = h[2 * k + 1];
                acc = __builtin_amdgcn_wmma_f32_16x16x4_f32(
                    false, av, false, bv, (short)0, acc, false, false);
            }

            // Layer 3: dy = sum_n relu(h2[n]) * W3[n]. Hidden index lives on
            // (reg, half): 8 in-lane FMAs + one cross-half add. ReLU on the
            // WMMA result via v_med3_f32(x, 0, +inf) — a single instruction,
            // avoiding fmaxf's NaN-canonicalization double-max.
            float s = 0.0f;
#pragma unroll
            for (int r = 0; r < 8; ++r)
                s = fmaf(__builtin_amdgcn_fmed3f(acc[r], 0.0f, inf), w3c[r], s);
            s += __shfl_xor(s, 16, 32);   // other half covers hidden 8..15 / 0..7

            // Euler update: y += dt * (dy + b3)
            yA[tau] = fmaf(dt, s + b3s, yA[tau]);
        }

        orow[lane] = (lane < 16) ? yA[0] : yA[1];
    }
}

extern "C" void kernel_launch(void* const* d_in, const int* in_sizes, int n_in,
                              void* d_out, int out_size, void* d_ws, size_t ws_size,
                              hipStream_t stream) {
    const float* y0 = (const float*)d_in[0];
    const float* t  = (const float*)d_in[1];
    const float* W1 = (const float*)d_in[2];
    const float* b1 = (const float*)d_in[3];
    const float* W2 = (const float*)d_in[4];
    const float* b2 = (const float*)d_in[5];
    const float* W3 = (const float*)d_in[6];
    const float* b3 = (const float*)d_in[7];
    float* out = (float*)d_out;

    const int B = in_sizes[0];   // 65536 trajectories
    const int T = in_sizes[1];   // 1000 time points

    const int threads = 256;                      // 8 waves; 32 trajectories/wave
    const int blocks  = B / (threads / 32 * 32);  // = B/256 = 256 blocks
    node_euler_wmma<<<blocks, threads, 0, stream>>>(y0, t, W1, b1, W2, b2,
                                                    W3, b3, out, B, T);
}